// LIF_R_weights_only_11347303596315
// MI455X (gfx1250) — compile-verified
//
#include <hip/hip_runtime.h>
#include <hip/hip_bf16.h>
#include <math.h>

// ---------------------------------------------------------------------------
// LIF_R soft-spike step, N=8192.
// Dominant cost: I_syn = (s*neuron_types) @ w  -> stream 256 MB of FP32 W.
// Bandwidth-bound (0.5 FLOP/byte); floor ~ 256MB / 23.3TB/s ~ 11 us.
// GEMV reduction is mapped onto V_WMMA_F32_16X16X4_F32 (free compute,
// full FP32 precision, exercises the CDNA5 matrix pipe).
// ---------------------------------------------------------------------------

#define NN 8192
#define COLS_PER_BLOCK 256              // 8 waves * 32 cols
#define ROWS_PER_CHUNK 256
#define NUM_CHUNKS   (NN / ROWS_PER_CHUNK)   // 32
#define COL_BLOCKS   (NN / COLS_PER_BLOCK)   // 32

typedef __attribute__((ext_vector_type(2))) float v2f;
typedef __attribute__((ext_vector_type(8))) float v8f;

// --- kernel 1: sv[i] = s[i] * neuron_types[i] ------------------------------
__global__ void lif_prep_sv(const float* __restrict__ s,
                            const float* __restrict__ nt,
                            float* __restrict__ sv) {
    int i = blockIdx.x * blockDim.x + threadIdx.x;
    if (i < NN) sv[i] = s[i] * nt[i];
}

// --- kernel 2: partial[chunk][j] = sum_{i in chunk} sv[i] * w[i][j] --------
// Each wave owns 32 columns (two 16-col WMMA tiles) and walks its row chunk
// in K-groups of 4, accumulating with V_WMMA_F32_16X16X4_F32.
//   A(16x4): A[m][k] = sv[i0+k] for all m  (broadcast over M)
//   B(4x16): B[k][n] = w[i0+k][j0+n]
//   D[m][n] = sum_k sv[i0+k]*w[i0+k][j0+n]  (all rows equal; read row 0)
__global__ void __launch_bounds__(256)
lif_gemv_wmma(const float* __restrict__ w,
              const float* __restrict__ sv,
              float* __restrict__ partial) {
    const int lane  = threadIdx.x & 31;
    const int wave  = threadIdx.x >> 5;                 // 0..7
    const int n     = lane & 15;                        // column within tile
    const int half  = lane >> 4;                        // 0: K=0,1  1: K=2,3
    const int j0    = blockIdx.x * COLS_PER_BLOCK + wave * 32;
    const int rowBase = blockIdx.y * ROWS_PER_CHUNK;

    v8f acc0 = {0.f, 0.f, 0.f, 0.f, 0.f, 0.f, 0.f, 0.f};
    v8f acc1 = {0.f, 0.f, 0.f, 0.f, 0.f, 0.f, 0.f, 0.f};

    for (int i0 = rowBase; i0 < rowBase + ROWS_PER_CHUNK; i0 += 4) {
        const int k0 = i0 + 2 * half;                   // row for VGPR0 of A/B
        const float* wrow0 = w + (size_t)k0 * NN;
        const float* wrow1 = wrow0 + NN;

        // A fragment: sv broadcast across M (per 16x4 f32 layout)
        v2f a;
        a.x = sv[k0];
        a.y = sv[k0 + 1];

        // prefetch a few rows ahead (emits global_prefetch_b8)
        __builtin_prefetch(wrow0 + (size_t)16 * NN + j0 + n, 0, 1);

        // tile 0: columns j0 .. j0+15
        v2f b0;
        b0.x = wrow0[j0 + n];
        b0.y = wrow1[j0 + n];
        acc0 = __builtin_amdgcn_wmma_f32_16x16x4_f32(
            false, a, false, b0, (short)0, acc0, false, false);

        // tile 1: columns j0+16 .. j0+31
        v2f b1;
        b1.x = wrow0[j0 + 16 + n];
        b1.y = wrow1[j0 + 16 + n];
        acc1 = __builtin_amdgcn_wmma_f32_16x16x4_f32(
            false, a, false, b1, (short)0, acc1, false, false);
    }

    // D row 0 lives in acc[0] of lanes 0..15
    if (lane < 16) {
        float* p = partial + (size_t)blockIdx.y * NN + j0;
        p[lane]      = acc0[0];
        p[16 + lane] = acc1[0];
    }
}

// --- kernel 3: reduce partials, diagonal fixup, LIF pointwise math ---------
__global__ void lif_finalize(const float* __restrict__ partial,
                             const float* __restrict__ sv,
                             const float* __restrict__ I_ext,
                             const float* __restrict__ w,
                             const float* __restrict__ s,
                             const float* __restrict__ v,
                             const float* __restrict__ theta_s,
                             const float* __restrict__ G,
                             const float* __restrict__ E_L,
                             const float* __restrict__ tau_m,
                             const float* __restrict__ b_s,
                             const float* __restrict__ delta_theta_s,
                             float* __restrict__ out) {
    int j = blockIdx.x * blockDim.x + threadIdx.x;
    if (j >= NN) return;

    float acc = 0.f;
    #pragma unroll 8
    for (int c = 0; c < NUM_CHUNKS; ++c)
        acc += partial[(size_t)c * NN + j];

    const float I_syn = acc - sv[j] * w[(size_t)j * NN + j];

    const float el  = E_L[j];
    const float th  = theta_s[j];
    const float norm_R = (delta_theta_s[j] - el) * 1.1f;

    const float dv     = (G[j] * (el - v[j]) + (I_syn + I_ext[j]) * norm_R) / tau_m[j];
    const float v_next = v[j] + dv;

    const float spiked    = (v_next >= th) ? 1.0f : 0.0f;
    const float theta_new = (1.0f - b_s[j]) * th + spiked * delta_theta_s[j];

    // soft_spiked = sigmoid(v_next - theta_new)
    const float x = v_next - theta_new;
    out[j] = 1.0f / (1.0f + __expf(-x));
    (void)s; // s only feeds discarded state updates in the reference output
}

extern "C" void kernel_launch(void* const* d_in, const int* in_sizes, int n_in,
                              void* d_out, int out_size, void* d_ws, size_t ws_size,
                              hipStream_t stream) {
    // setup_inputs() order:
    const float* I_ext         = (const float*)d_in[0];
    const float* w             = (const float*)d_in[1];
    const float* s             = (const float*)d_in[2];
    const float* v             = (const float*)d_in[3];
    const float* theta_s       = (const float*)d_in[4];
    const float* neuron_types  = (const float*)d_in[5];
    const float* G             = (const float*)d_in[6];
    const float* E_L           = (const float*)d_in[7];
    const float* tau_m         = (const float*)d_in[8];
    const float* tau_s         = (const float*)d_in[9];   // unused for output
    const float* b_s           = (const float*)d_in[10];
    const float* delta_theta_s = (const float*)d_in[11];
    // d_in[12] f_v, d_in[13] delta_V: dead w.r.t. returned soft_spiked
    (void)tau_s; (void)in_sizes; (void)n_in; (void)out_size; (void)ws_size;

    float* sv      = (float*)d_ws;                    // 8192 floats
    float* partial = sv + NN;                         // 32 * 8192 floats

    lif_prep_sv<<<NN / 256, 256, 0, stream>>>(s, neuron_types, sv);

    dim3 grid(COL_BLOCKS, NUM_CHUNKS);
    lif_gemv_wmma<<<grid, 256, 0, stream>>>(w, sv, partial);

    lif_finalize<<<NN / 256, 256, 0, stream>>>(partial, sv, I_ext, w, s, v,
                                               theta_s, G, E_L, tau_m,
                                               b_s, delta_theta_s,
                                               (float*)d_out);
}